// VQWeightDehazeNet_DQA_7481833029773
// MI455X (gfx1250) — compile-verified
//
#include <hip/hip_runtime.h>
#include <hip/hip_bf16.h>
#include <math.h>

// ---------------------------------------------------------------------------
// VQ nearest-code (argmin over ||e||^2 - 2 x.e) for z and gt.
// Fused GEMM+argmin on V_WMMA_F32_16X16X4_F32; codebook + x tiles staged into
// LDS by the Tensor Data Mover (tensor_load_to_lds, TENSORcnt), double-buffered.
// Then gather + cosine-over-H reduction.
// ---------------------------------------------------------------------------

typedef __attribute__((ext_vector_type(2))) float v2f;
typedef __attribute__((ext_vector_type(4))) float v4f;
typedef __attribute__((ext_vector_type(8))) float v8f;
typedef __attribute__((ext_vector_type(4))) unsigned int u32x4;
typedef __attribute__((ext_vector_type(4))) int i32x4;
typedef __attribute__((ext_vector_type(8))) int i32x8;

#define NB    64
#define CDIM  256
#define HH    32
#define WW    32
#define NTOK  (NB * HH * WW)          // 65536 flattened vectors
#define NE    1024                    // codebook entries
#define EPSV  1e-8f

// d_out layout (floats), reference return order:
//   zq [NTOK*CDIM] | cosine [NB*WW*CDIM] | idx_gt [NTOK] | idx [NTOK]
#define ZQ_N      (NTOK * CDIM)
#define COS_OFF   (ZQ_N)
#define COS_N     (NB * WW * CDIM)
#define IDXGT_OFF (COS_OFF + COS_N)
#define IDX_OFF   (IDXGT_OFF + NTOK)

#define CB_LDSTRIDE 260   // 256 data dwords + 4 pad dwords per codebook row

#if __has_builtin(__builtin_amdgcn_tensor_load_to_lds)
#define USE_TDM 1
#else
#define USE_TDM 0
#endif

#if USE_TDM
// 2D tensor_load_to_lds: tile_d0 contiguous elements per row, tile_d1 rows,
// row stride stride0 (4-byte elements). w0flags carries data_size + padding.
__device__ __forceinline__ void tdm_load_2d(unsigned lds_off, const float* gptr,
                                            unsigned tensor_d0, unsigned tensor_d1,
                                            unsigned tile_d0, unsigned tile_d1,
                                            unsigned stride0, unsigned w0flags) {
    unsigned long long ga = (unsigned long long)(uintptr_t)gptr;
    u32x4 g0;
    g0[0] = 1u;                                    // count=1, load, not restore
    g0[1] = lds_off;                               // LDS byte address
    g0[2] = (unsigned)ga;                          // global_addr[31:0]
    g0[3] = (unsigned)((ga >> 32) & 0x1FFFFFFu) | (2u << 30);  // addr[56:32] | type=2
    i32x8 g1;
    g1[0] = (int)w0flags;                          // data_size / pad config
    g1[1] = (int)((tensor_d0 & 0xFFFFu) << 16);    // tensor_dim0[15:0]
    g1[2] = (int)(((tensor_d0 >> 16) & 0xFFFFu) | ((tensor_d1 & 0xFFFFu) << 16));
    g1[3] = (int)(((tensor_d1 >> 16) & 0xFFFFu) | ((tile_d0 & 0xFFFFu) << 16));
    g1[4] = (int)(tile_d1 & 0xFFFFu);              // tile_dim1 (tile_dim2=0)
    g1[5] = (int)stride0;                          // tensor_dim0_stride[31:0]
    g1[6] = 0;                                     // stride hi / dim1_stride lo
    g1[7] = 0;
    i32x4 gz = {0, 0, 0, 0};
#if defined(__clang_major__) && __clang_major__ >= 23
    i32x8 gz8 = {0, 0, 0, 0, 0, 0, 0, 0};
    __builtin_amdgcn_tensor_load_to_lds(g0, g1, gz, gz, gz8, 0);
#else
    __builtin_amdgcn_tensor_load_to_lds(g0, g1, gz, gz, 0);
#endif
}
// data_size=4B | pad_enable | pad_interval=256dw | pad_amount=4dw
#define CB_TDM_FLAGS ((2u << 16) | (1u << 20) | (7u << 22) | (3u << 25))
#define X_TDM_FLAGS  (2u << 16)
#endif

// ---------------------------------------------------------------------------
// Kernel 1: codebook squared norms -> d_ws (4 KB)
// ---------------------------------------------------------------------------
__global__ void vq_norms(const float* __restrict__ cb, float* __restrict__ norms) {
    int n = blockIdx.x * blockDim.x + threadIdx.x;
    if (n >= NE) return;
    const v4f* p = (const v4f*)(cb + n * CDIM);
    float s = 0.f;
#pragma unroll 8
    for (int i = 0; i < CDIM / 4; ++i) {
        v4f v = p[i];
        s += v.x * v.x + v.y * v.y + v.z * v.z + v.w * v.w;
    }
    norms[n] = s;
}

// ---------------------------------------------------------------------------
// Kernel 2: fused distance-GEMM + argmin.
// grid = (NTOK/128, 2); block = 256 (8 waves). Each wave owns one 16-row
// M-tile; the 16-code column tile is TDM-staged in LDS and shared by all
// 8 waves (8x less L2 codebook traffic than 16-row blocks).
// ---------------------------------------------------------------------------
__launch_bounds__(256)
__global__ void vq_argmin(const float* __restrict__ z,
                          const float* __restrict__ gt,
                          const float* __restrict__ cb,
                          const float* __restrict__ norms,
                          float* __restrict__ out) {
    __shared__ float xb[8][16 * CDIM];            // per-wave x tile, [c][m]  (128 KB)
    __shared__ float cbuf[2][16 * CB_LDSTRIDE];   // padded codebook tiles    (32.5 KB)
    __shared__ float n_lds[NE];                   // 4 KB

    const int tid   = threadIdx.x;
    const int wave  = tid >> 5;
    const int lane  = tid & 31;
    const int which = blockIdx.y;                 // 0 -> z, 1 -> gt
    const float* __restrict__ src = which ? gt : z;

    const int gn0  = blockIdx.x * 128;            // first row of this block
    const int bbat = gn0 >> 10;                   // batch (128 | 1024, never straddles)
    const int hw0  = gn0 & 1023;
    const size_t base = (size_t)bbat * (CDIM * HH * WW) + hw0;

#if USE_TDM
    // Per-wave x tile: 16 contiguous floats per channel row, stride HW=1024;
    // lands in LDS exactly column-major [c*16+m].
    tdm_load_2d((unsigned)(uintptr_t)&xb[wave][0], src + base + wave * 16,
                16, CDIM, 16, CDIM, HH * WW, X_TDM_FLAGS);
    if (wave == 0)   // prefetch codebook tile 0 (padded rows, stride 260 dw)
        tdm_load_2d((unsigned)(uintptr_t)&cbuf[0][0], cb,
                    CDIM, 16, CDIM, 16, CDIM, CB_TDM_FLAGS);
#else
    for (int i = 0; i < 128; ++i) {               // cooperative fallback staging
        int lin = i * 256 + tid;
        int m128 = lin & 127, c = lin >> 7;
        xb[m128 >> 4][c * 16 + (m128 & 15)] = src[base + (size_t)c * (HH * WW) + m128];
    }
#endif
#pragma unroll
    for (int j = 0; j < NE / 256; ++j)
        n_lds[j * 256 + tid] = norms[j * 256 + tid];
#if USE_TDM
    __builtin_amdgcn_s_wait_tensorcnt(0);         // own x tile (+ cb tile on wave 0)
#endif
    __syncthreads();

    const int m     = lane & 15;                  // D column within tile
    const int khalf = (lane >> 4) << 1;           // K sub-offset (f32 A/B layout)
    const float* __restrict__ xw = &xb[wave][0];

    float best[8];
    int   bidx[8];
#pragma unroll
    for (int r = 0; r < 8; ++r) { best[r] = __builtin_inff(); bidx[r] = 0x7fffffff; }

    for (int ct = 0; ct < 64; ++ct) {
#if USE_TDM
        const int cur = ct & 1;
        if (wave == 0 && ct < 63)                 // DMA next tile over this compute
            tdm_load_2d((unsigned)(uintptr_t)&cbuf[cur ^ 1][0],
                        cb + (size_t)(ct + 1) * 16 * CDIM,
                        CDIM, 16, CDIM, 16, CDIM, CB_TDM_FLAGS);
#else
        const int cur = 0;
        __syncthreads();
        for (int i = 0; i < 16; ++i) {
            int lin = i * 256 + tid;
            int r = lin >> 8, k = lin & 255;
            cbuf[0][r * CB_LDSTRIDE + k] = cb[(size_t)(ct * 16 + r) * CDIM + k];
        }
        __syncthreads();
#endif
        const float* __restrict__ cbl = &cbuf[cur][m * CB_LDSTRIDE + khalf];

        v8f accA = {0.f, 0.f, 0.f, 0.f, 0.f, 0.f, 0.f, 0.f};
        v8f accB = {0.f, 0.f, 0.f, 0.f, 0.f, 0.f, 0.f, 0.f};
#pragma unroll 4
        for (int kc = 0; kc < 64; kc += 2) {      // dual accumulators: no WMMA RAW chain
            const int k0 = kc * 4;
            v2f a0;
            a0.x = xw[(k0 + khalf)     * 16 + m];
            a0.y = xw[(k0 + khalf + 1) * 16 + m];
            v2f b0 = *(const v2f*)(cbl + k0);     // 8B-aligned b64, banks 4n+k: no conflict
            accA = __builtin_amdgcn_wmma_f32_16x16x4_f32(
                       false, a0, false, b0, (short)0, accA, false, false);
            const int k1 = k0 + 4;
            v2f a1;
            a1.x = xw[(k1 + khalf)     * 16 + m];
            a1.y = xw[(k1 + khalf + 1) * 16 + m];
            v2f b1 = *(const v2f*)(cbl + k1);
            accB = __builtin_amdgcn_wmma_f32_16x16x4_f32(
                       false, a1, false, b1, (short)0, accB, false, false);
        }

        const int n  = ct * 16 + m;               // code id of this lane's column
        const float nv = n_lds[n];
#pragma unroll
        for (int r = 0; r < 8; ++r) {             // d = ||e||^2 - 2 x.e
            float d = nv - 2.0f * (accA[r] + accB[r]);
            if (d < best[r] || (d == best[r] && n < bidx[r])) { best[r] = d; bidx[r] = n; }
        }
#if USE_TDM
        if (wave == 0) __builtin_amdgcn_s_wait_tensorcnt(0);
        __syncthreads();                          // next tile resident before anyone reads
#endif
    }

    // Reduce across the 16 lanes of each half-wave (rows 0-7 live in lanes
    // 0-15, rows 8-15 in lanes 16-31 per the f32 C/D VGPR layout).
#pragma unroll
    for (int r = 0; r < 8; ++r) {
#pragma unroll
        for (int off = 8; off >= 1; off >>= 1) {
            float ob = __shfl_xor(best[r], off, 16);
            int   oi = __shfl_xor(bidx[r], off, 16);
            if (ob < best[r] || (ob == best[r] && oi < bidx[r])) { best[r] = ob; bidx[r] = oi; }
        }
    }
    if ((lane & 15) == 0) {                       // lanes 0 and 16 hold final rows
        const int half = lane >> 4;
        const int obase = (which ? IDXGT_OFF : IDX_OFF) + gn0 + wave * 16 + half * 8;
#pragma unroll
        for (int r = 0; r < 8; ++r)
            out[obase + r] = (float)bidx[r];      // exact: idx < 1024
    }
}

// ---------------------------------------------------------------------------
// Kernel 3: zq = codebook[idx]  (flattened [N,C] == (B,H,W,C) output layout)
// ---------------------------------------------------------------------------
__global__ void vq_gather(const float* __restrict__ cb, float* __restrict__ out) {
    const int n = blockIdx.x;
    const int c = threadIdx.x;
    const int iz = (int)out[IDX_OFF + n];
    out[(size_t)n * CDIM + c] = cb[iz * CDIM + c];
}

// ---------------------------------------------------------------------------
// Kernel 4: cosine over H axis -> [B, W, C]
// ---------------------------------------------------------------------------
__global__ void vq_cosine(const float* __restrict__ cb, float* __restrict__ out) {
    __shared__ int iz[HH], ig[HH];
    const int bw = blockIdx.x;
    const int b  = bw >> 5;
    const int w  = bw & 31;
    const int tid = threadIdx.x;                  // 256 = CDIM

    if (tid < HH) {
        const int n = (b * HH + tid) * WW + w;
        iz[tid] = (int)out[IDX_OFF   + n];
        ig[tid] = (int)out[IDXGT_OFF + n];
    }
    __syncthreads();

    float num = 0.f, nz = 0.f, ng = 0.f;
#pragma unroll 4
    for (int h = 0; h < HH; ++h) {
        float a = cb[iz[h] * CDIM + tid];
        float g = cb[ig[h] * CDIM + tid];
        num += g * a; nz += a * a; ng += g * g;
    }
    float den = fmaxf(sqrtf(ng), EPSV) * fmaxf(sqrtf(nz), EPSV);
    out[COS_OFF + (b * WW + w) * CDIM + tid] = num / den;
}

// ---------------------------------------------------------------------------
extern "C" void kernel_launch(void* const* d_in, const int* in_sizes, int n_in,
                              void* d_out, int out_size, void* d_ws, size_t ws_size,
                              hipStream_t stream) {
    const float* z  = (const float*)d_in[0];
    const float* gt = (const float*)d_in[1];
    const float* cb = (const float*)d_in[2];
    float* out   = (float*)d_out;
    float* norms = (float*)d_ws;                  // 4 KB scratch

    vq_norms<<<NE / 256, 256, 0, stream>>>(cb, norms);

    dim3 g2(NTOK / 128, 2);
    vq_argmin<<<g2, 256, 0, stream>>>(z, gt, cb, norms, out);

    vq_gather<<<NTOK, CDIM, 0, stream>>>(cb, out);
    vq_cosine<<<NB * WW, CDIM, 0, stream>>>(cb, out);
}